// TreeCRFVanilla_11184094838935
// MI455X (gfx1250) — compile-verified
//
#include <hip/hip_runtime.h>
#include <math.h>

// TreeCRF inside (CKY) partition for B=32, N=128, L=64, fp32.
//
// Roofline: stage 1 streams 134MB (HBM-bound, ~6us floor @23.3TB/s).
// Stage 2 is a 127-deep serial chain per batch; we run it entirely in LDS
// (64KB s-table out of 320KB/WGP), use V_WMMA_F32_16X16X4_F32 in the exp
// domain for the anti-diagonal k-reductions, and hide slp fetch latency with
// GLOBAL_LOAD_ASYNC_TO_LDS double buffering (ASYNCcnt).

#define BB 32
#define NN 128
#define LL 64

typedef __attribute__((ext_vector_type(2))) float v2f;
typedef __attribute__((ext_vector_type(8))) float v8f;

#define NEG_INF (-__builtin_inff())

// ---- CDNA5 async global->LDS copy (ASYNCcnt path), GV addressing mode ----
__device__ __forceinline__ void async_ld_b32(unsigned lds_byte_addr, const float* g) {
  asm volatile("global_load_async_to_lds_b32 %0, %1, off"
               :: "v"(lds_byte_addr), "v"((unsigned long long)(size_t)g)
               : "memory");
}
__device__ __forceinline__ void wait_async0() {
  asm volatile("s_wait_asynccnt 0x0" ::: "memory");
}

__device__ __forceinline__ float v8_sel(v8f a, int s) {
  float r = a[0];
  r = (s == 1) ? a[1] : r;
  r = (s == 2) ? a[2] : r;
  r = (s == 3) ? a[3] : r;
  r = (s == 4) ? a[4] : r;
  r = (s == 5) ? a[5] : r;
  r = (s == 6) ? a[6] : r;
  r = (s == 7) ? a[7] : r;
  return r;
}

// ================= Stage 1: slp[b,i,j] = logsumexp_l lp[b,i,j,l] =================
__global__ __launch_bounds__(256) void lse_labels_kernel(const float* __restrict__ lp,
                                                         float* __restrict__ slp) {
  int cell = blockIdx.x * 256 + threadIdx.x;      // grid exactly covers B*N*N
  const float4* p = (const float4*)(lp + (size_t)cell * LL);
  float m = NEG_INF, ssum = 0.f;
#pragma unroll
  for (int q = 0; q < LL / 4; ++q) {
    float4 v = p[q];
    float m4 = fmaxf(fmaxf(v.x, v.y), fmaxf(v.z, v.w));
    float nm = fmaxf(m, m4);
    float e = __expf(v.x - nm) + __expf(v.y - nm) + __expf(v.z - nm) + __expf(v.w - nm);
    ssum = (q ? ssum * __expf(m - nm) : 0.f) + e;
    m = nm;
  }
  slp[cell] = m + __logf(ssum);
}

// ================= Stage 2: CKY diagonals in LDS, one WG per batch =================
__global__ __launch_bounds__(1024) void inside_kernel(const float* __restrict__ slp,
                                                      float* __restrict__ out) {
  extern __shared__ float smem[];
  float* sL   = smem;               // NN*NN : s table (row-major, stride NN)
  float* rmax = smem + NN * NN;     // NN    : running row max of s
  float* cmax = rmax + NN;          // NN    : running col max of s
  float* pbuf = cmax + NN;          // 2*NN  : slp anti-diagonal double buffer

  const int tid  = threadIdx.x;
  const int lane = tid & 31;
  const int wv   = tid >> 5;        // wave id, 0..31
  const int b    = blockIdx.x;
  const float* slp_b = slp + (size_t)b * NN * NN;

  // ---- init: d=0 diagonal + running maxes ----
  for (int c = tid; c < NN; c += 1024) {
    float v = slp_b[c * (NN + 1)];
    sL[c * NN + c] = v;
    rmax[c] = v;
    cmax[c] = v;
  }
  // prefetch slp diagonal d=1 into pbuf[1] (every wave issues; waits on its own cnt)
  {
    float* dst = pbuf + NN;
    for (int r = lane; r < NN - 1; r += 32)
      async_ld_b32((unsigned)(size_t)&dst[r], slp_b + r * (NN + 1) + 1);
  }
  wait_async0();
  __syncthreads();

  for (int d = 1; d < NN; ++d) {
    const int cur = d & 1;
    const int C = NN - d;
    float* bufc = pbuf + NN * cur;

    // prefetch next diagonal's slp while we compute this one
    if (d + 1 < NN) {
      float* dst = pbuf + NN * (cur ^ 1);
      for (int r = lane; r < C - 1; r += 32)
        async_ld_b32((unsigned)(size_t)&dst[r], slp_b + r * (NN + 1) + d + 1);
    }

    if (d <= 8) {
      // ---- R0: lane per cell, serial online LSE over k ----
      if (tid < C) {
        const int i = tid;
        const int la = i * (NN + 1);
        const int ra = (i + 1) * NN + i + d;
        float m = NEG_INF, ss = 0.f;
        for (int k = 0; k < d; ++k) {
          float v = sL[la + k] + sL[ra + (k << 7)];
          float nm = fmaxf(m, v);
          ss = ((m > NEG_INF) ? ss * __expf(m - nm) : 0.f) + __expf(v - nm);
          m = nm;
        }
        float o = m + __logf(ss) + bufc[i];
        sL[i * NN + i + d] = o;
        rmax[i] = fmaxf(rmax[i], o);
        cmax[i + d] = fmaxf(cmax[i + d], o);
      }
    } else {
      const int ntiles = C >> 4;          // 16-cell WMMA tiles (0 when C<16)
      const int nrem   = C - (ntiles << 4);
      if (wv < ntiles) {
        // ---- R1: exp-domain reduction on the matrix pipe ----
        // out_i = m_i + log( sum_k exp(L[i,k]+R[k,i]-m_i) ),
        // m_i = rowmax_i + colmax_{i+d}  (safe upper bound; exp terms <= 1).
        // A(16x4 f32): lane l holds M=l&15; lanes<16 -> K={4q,4q+1}, lanes>=16 -> K={4q+2,4q+3}.
        // B = ones (layout invariant)  => D[M][*] = row sum, replicated over columns.
        const int T = wv << 4;
        const int i = T + (lane & 15);
        const float shift = rmax[i] + cmax[i + d];
        const int kh = (lane >> 4) << 1;
        const int la = i * (NN + 1);
        const int ra = (i + 1) * NN + i + d;
        v2f ones; ones.x = 1.0f; ones.y = 1.0f;
        v8f acc = {0.f, 0.f, 0.f, 0.f, 0.f, 0.f, 0.f, 0.f};
        const int nq = d >> 2;
        for (int q = 0; q < nq; ++q) {
          int k = (q << 2) + kh;
          float t0 = sL[la + k]     + sL[ra + (k << 7)];
          float t1 = sL[la + k + 1] + sL[ra + ((k + 1) << 7)];
          v2f a;
          a.x = __expf(t0 - shift);
          a.y = __expf(t1 - shift);
          acc = __builtin_amdgcn_wmma_f32_16x16x4_f32(false, a, false, ones,
                                                      (short)0, acc, false, false);
        }
        // extraction: D VGPR r holds M=r (lanes 0-15) / M=r+8 (lanes 16-31)
        if ((lane < 8) || (lane >= 16 && lane < 24)) {
          const int m8 = (lane & 7) + ((lane >> 4) << 3);
          const int ci = T + m8;
          const float sh2 = rmax[ci] + cmax[ci + d];
          float sum = v8_sel(acc, lane & 7);
          const int la2 = ci * (NN + 1);
          const int ra2 = (ci + 1) * NN + ci + d;
          for (int k = nq << 2; k < d; ++k)   // k remainder (d%4)
            sum += __expf(sL[la2 + k] + sL[ra2 + (k << 7)] - sh2);
          float o = sh2 + __logf(sum) + bufc[ci];
          sL[ci * NN + ci + d] = o;
          rmax[ci] = fmaxf(rmax[ci], o);
          cmax[ci + d] = fmaxf(cmax[ci + d], o);
        }
      } else if (wv < ntiles + nrem) {
        // ---- R2: wave per leftover cell, lanes split k, shuffle LSE reduce ----
        const int i = (ntiles << 4) + (wv - ntiles);
        const int la = i * (NN + 1);
        const int ra = (i + 1) * NN + i + d;
        float m = NEG_INF, ss = 0.f;
        for (int k = lane; k < d; k += 32) {
          float v = sL[la + k] + sL[ra + (k << 7)];
          float nm = fmaxf(m, v);
          ss = ((m > NEG_INF) ? ss * __expf(m - nm) : 0.f) + __expf(v - nm);
          m = nm;
        }
#pragma unroll
        for (int off = 16; off; off >>= 1) {
          float m2 = __shfl_xor(m, off, 32);
          float s2 = __shfl_xor(ss, off, 32);
          float nm = fmaxf(m, m2);
          float p1 = (m  > NEG_INF) ? ss * __expf(m  - nm) : 0.f;
          float p2 = (m2 > NEG_INF) ? s2 * __expf(m2 - nm) : 0.f;
          m = nm; ss = p1 + p2;
        }
        if (lane == 0) {
          float o = m + __logf(ss) + bufc[i];
          sL[i * NN + i + d] = o;
          rmax[i] = fmaxf(rmax[i], o);
          cmax[i + d] = fmaxf(cmax[i + d], o);
        }
      }
    }

    wait_async0();      // this wave's slp prefetch for d+1 has landed in LDS
    __syncthreads();    // single barrier per diagonal
  }

  if (tid == 0) out[b] = sL[NN - 1];   // s[0][N-1] == log Z
}

extern "C" void kernel_launch(void* const* d_in, const int* in_sizes, int n_in,
                              void* d_out, int out_size, void* d_ws, size_t ws_size,
                              hipStream_t stream) {
  (void)in_sizes; (void)n_in; (void)out_size; (void)ws_size;
  const float* lp = (const float*)d_in[0];
  float* slp = (float*)d_ws;              // B*N*N floats = 2MB scratch
  float* out = (float*)d_out;

  const int total = BB * NN * NN;         // 524288 cells
  lse_labels_kernel<<<dim3(total / 256), dim3(256), 0, stream>>>(lp, slp);

  const size_t shmem = (size_t)(NN * NN + 4 * NN) * sizeof(float);  // 67584 B
  inside_kernel<<<dim3(BB), dim3(1024), shmem, stream>>>(slp, out);
}